// ResGCNModel_5153960755351
// MI455X (gfx1250) — compile-verified
//
#include <hip/hip_runtime.h>
#include <hip/hip_bf16.h>

// ResGCN (2-layer GCNConv + residual), N=100000 nodes, d=64, E=1.6M edges.
// All tensors fit in the MI455X 192MB L2, so the edge scatter/gather runs at
// L2 bandwidth; GEMMs use full-fp32 V_WMMA_F32_16X16X4_F32 (precision-exact
// vs reference, and the GEMM is ~0.8 GFLOP -> negligible either way).

typedef float v2f __attribute__((ext_vector_type(2)));
typedef float v8f __attribute__((ext_vector_type(8)));

#define LANES 32

// ---------- degree / norm kernels ----------

__global__ __launch_bounds__(256) void k_set1(float* __restrict__ d, int n) {
    int i = blockIdx.x * 256 + threadIdx.x;
    if (i < n) d[i] = 1.0f;  // self-loop contributes 1 to every node's degree
}

__global__ __launch_bounds__(256) void k_degree(const long long* __restrict__ ei,
                                                float* __restrict__ deg, int E) {
    int e = blockIdx.x * 256 + threadIdx.x;
    if (e < E) {
        long long dst = ei[(size_t)E + e];
        unsafeAtomicAdd(&deg[dst], 1.0f);  // native global_atomic_add_f32
    }
}

__global__ __launch_bounds__(256) void k_rsqrt(float* __restrict__ d, int n) {
    int i = blockIdx.x * 256 + threadIdx.x;
    if (i < n) d[i] = rsqrtf(d[i]);  // deg >= 1 always (self-loops)
}

// ---------- WMMA fp32 GEMM: H[N,64] = act(X[N,64]) @ W[64,64] ----------
// One wave per 16x64 output tile. B (the 64x64 weight) is hoisted into
// registers as 4x16 v2f fragments; A streams per tile; accumulate with
// v_wmma_f32_16x16x4_f32 (16 k-steps per 16-wide n-tile).

__global__ __launch_bounds__(256) void gcn_gemm_wmma(const float* __restrict__ X,
                                                     const float* __restrict__ W,
                                                     float* __restrict__ H,
                                                     int nTiles, int relu) {
    const int lane = threadIdx.x & (LANES - 1);
    const int wave = threadIdx.x / LANES;
    const int half = lane >> 4;   // 0: lanes 0-15, 1: lanes 16-31
    const int mn   = lane & 15;

    int tile = blockIdx.x * 8 + wave;
    if (tile >= nTiles) return;
    const int tileStride = gridDim.x * 8;

    // Hoist all B fragments (whole 64x64 weight) into registers.
    // Layout per 16x16x4 fp32 WMMA: B is 4x16 (KxN); lane holds column
    // N = mn; VGPR0 = row K0+2*half, VGPR1 = row K0+2*half+1.
    v2f Bf[4][16];
#pragma unroll
    for (int j = 0; j < 4; ++j) {
#pragma unroll
        for (int s = 0; s < 16; ++s) {
            const int k = 4 * s + 2 * half;
            const int n = j * 16 + mn;
            Bf[j][s].x = W[(size_t)k * 64 + n];
            Bf[j][s].y = W[(size_t)(k + 1) * 64 + n];
        }
    }

    do {
        const int row0 = tile * 16;

        // A fragments: 16x4 (MxK); lane holds row M = mn;
        // VGPR0 = K0+2*half, VGPR1 = K0+2*half+1. ReLU fused on load.
        v2f Af[16];
#pragma unroll
        for (int s = 0; s < 16; ++s) {
            const int k = 4 * s + 2 * half;
            const size_t base = (size_t)(row0 + mn) * 64 + k;
            float a0 = X[base];
            float a1 = X[base + 1];
            if (relu) { a0 = fmaxf(a0, 0.0f); a1 = fmaxf(a1, 0.0f); }
            Af[s].x = a0;
            Af[s].y = a1;
        }

#pragma unroll
        for (int j = 0; j < 4; ++j) {
            v8f c = {0.f, 0.f, 0.f, 0.f, 0.f, 0.f, 0.f, 0.f};
#pragma unroll
            for (int s = 0; s < 16; ++s) {
                c = __builtin_amdgcn_wmma_f32_16x16x4_f32(
                        false, Af[s], false, Bf[j][s], (short)0, c, false, false);
            }
            // D layout: VGPR r -> M = r + 8*half, N = mn.
#pragma unroll
            for (int r = 0; r < 8; ++r) {
                H[(size_t)(row0 + half * 8 + r) * 64 + j * 16 + mn] = c[r];
            }
        }
        tile += tileStride;
    } while (tile < nTiles);
}

// ---------- init: out = b + dinv^2 * h  (+ optional residual x) ----------
// Handles the self-loop edge (norm = dinv[i]*dinv[i]) and the bias, so the
// scatter kernel only processes the 1.6M real edges.

__global__ __launch_bounds__(256) void k_init_out(const float* __restrict__ h,
                                                  const float* __restrict__ dinv,
                                                  const float* __restrict__ b,
                                                  const float* __restrict__ xres,
                                                  float* __restrict__ out, int n) {
    int i = blockIdx.x * 256 + threadIdx.x;
    if (i >= n) return;
    const int node = i >> 6;
    const int f = i & 63;
    const float di = dinv[node];
    float v = b[f] + di * di * h[i];
    if (xres) v += xres[i];
    out[i] = v;
}

// ---------- edge scatter: out[dst] += dinv[src]*dinv[dst] * h[src] ----------
// 16 threads per edge, each handling a float4 feature chunk: 128-bit gather
// of h[src] (L2-resident) + 4 native fp32 atomics into out[dst].

__global__ __launch_bounds__(256) void k_scatter(const long long* __restrict__ ei,
                                                 const float* __restrict__ dinv,
                                                 const float* __restrict__ h,
                                                 float* __restrict__ out, int E) {
    const int t = blockIdx.x * 256 + threadIdx.x;
    const int e = t >> 4;
    if (e >= E) return;
    const int chunk = t & 15;

    const long long s = ei[e];
    const long long d = ei[(size_t)E + e];
    const float norm = dinv[s] * dinv[d];

    const float4 hv = *((const float4*)(h + ((size_t)s << 6)) + chunk);
    float* op = out + ((size_t)d << 6) + (chunk << 2);
    unsafeAtomicAdd(op + 0, norm * hv.x);
    unsafeAtomicAdd(op + 1, norm * hv.y);
    unsafeAtomicAdd(op + 2, norm * hv.z);
    unsafeAtomicAdd(op + 3, norm * hv.w);
}

// ---------- host launcher ----------

extern "C" void kernel_launch(void* const* d_in, const int* in_sizes, int n_in,
                              void* d_out, int out_size, void* d_ws, size_t ws_size,
                              hipStream_t stream) {
    const float*      x  = (const float*)d_in[0];
    const long long*  ei = (const long long*)d_in[1];   // int64 [2,E]
    const float*      W1 = (const float*)d_in[2];
    const float*      b1 = (const float*)d_in[3];
    const float*      W2 = (const float*)d_in[4];
    const float*      b2 = (const float*)d_in[5];
    float*            out = (float*)d_out;

    const int N  = in_sizes[0] / 64;   // 100000
    const int E  = in_sizes[1] / 2;    // 1600000
    const int NF = N * 64;

    // Workspace layout (~49.3 MiB total): dinv[N] | h[N*64] | tmp[N*64]
    char* ws = (char*)d_ws;
    float* dinv = (float*)ws;
    size_t off = (((size_t)N * 4 + 255) / 256) * 256;
    float* bufA = (float*)(ws + off);                         // h1 / h2
    off += (((size_t)NF * 4 + 255) / 256) * 256;
    float* bufB = (float*)(ws + off);                         // out1 / x1

    const int gN   = (N + 255) / 256;
    const int gE   = (E + 255) / 256;
    const int gNF  = (NF + 255) / 256;
    const long long scatterThreads = (long long)E * 16;
    const int gScat = (int)((scatterThreads + 255) / 256);
    const int nTiles = N / 16;                 // 100000/16 = 6250 exactly
    const int gGemm  = (nTiles + 7) / 8;       // 8 waves (tiles) per 256-thread block

    // degree -> dinv
    k_set1  <<<gN, 256, 0, stream>>>(dinv, N);
    k_degree<<<gE, 256, 0, stream>>>(ei, dinv, E);
    k_rsqrt <<<gN, 256, 0, stream>>>(dinv, N);

    // layer 1: h1 = x @ W1 ; out1 = b1 + dinv^2*h1 ; scatter edges
    gcn_gemm_wmma<<<gGemm, 256, 0, stream>>>(x, W1, bufA, nTiles, 0);
    k_init_out   <<<gNF, 256, 0, stream>>>(bufA, dinv, b1, nullptr, bufB, NF);
    k_scatter    <<<gScat, 256, 0, stream>>>(ei, dinv, bufA, bufB, E);

    // layer 2: h2 = relu(out1) @ W2 ; out = b2 + dinv^2*h2 + x ; scatter edges
    gcn_gemm_wmma<<<gGemm, 256, 0, stream>>>(bufB, W2, bufA, nTiles, 1);
    k_init_out   <<<gNF, 256, 0, stream>>>(bufA, dinv, b2, x, out, NF);
    k_scatter    <<<gScat, 256, 0, stream>>>(ei, dinv, bufA, out, E);
}